// ModalityBranchRN_12979391168872
// MI455X (gfx1250) — compile-verified
//
#include <hip/hip_runtime.h>
#include <math.h>

#define LEAKY 0.2f

typedef __attribute__((ext_vector_type(2))) float v2f;
typedef __attribute__((ext_vector_type(8))) float v8f;

__device__ __forceinline__ float leaky_f(float x) { return fmaxf(x, LEAKY * x); }

// ---------------------------------------------------------------------------
// Kernel 1: conv1d(1->8, k=5, pad=2) + leaky + maxpool4
// x: (128, 4096)  ->  h1: (128, 8, 1024)
// ---------------------------------------------------------------------------
__global__ void __launch_bounds__(256) conv1_pool4(const float* __restrict__ x,
                                                   const float* __restrict__ w,
                                                   const float* __restrict__ bias,
                                                   float* __restrict__ h1) {
  int idx = blockIdx.x * blockDim.x + threadIdx.x;   // 128*8*1024 total
  int n = idx & 1023;
  int c = (idx >> 10) & 7;
  int b = idx >> 13;
  const float* xb = x + b * 4096;
  float bb = bias[c];
  float m = -3.4e38f;
#pragma unroll
  for (int p = 0; p < 4; ++p) {
    int base = n * 4 + p - 2;
    float s = bb;
#pragma unroll
    for (int t = 0; t < 5; ++t) {
      int j = base + t;
      float xv = (j >= 0 && j < 4096) ? xb[j] : 0.0f;
      s = fmaf(xv, w[c * 5 + t], s);
    }
    m = fmaxf(m, leaky_f(s));
  }
  h1[idx] = m;
}

// ---------------------------------------------------------------------------
// Kernel 2: conv1d(8->8, k=5, pad=2) + leaky + maxpool8
// h1: (128, 8, 1024)  ->  h2: (128, 8, 128)
// ---------------------------------------------------------------------------
__global__ void __launch_bounds__(256) conv2_pool8(const float* __restrict__ h1,
                                                   const float* __restrict__ w,
                                                   const float* __restrict__ bias,
                                                   float* __restrict__ h2) {
  int idx = blockIdx.x * blockDim.x + threadIdx.x;   // 128*8*128 total
  int n = idx & 127;
  int c = (idx >> 7) & 7;
  int b = idx >> 10;
  const float* hb = h1 + b * 8 * 1024;
  float bb = bias[c];
  float m = -3.4e38f;
  for (int p = 0; p < 8; ++p) {
    int base = n * 8 + p - 2;
    float s = bb;
#pragma unroll
    for (int ci = 0; ci < 8; ++ci) {
      const float* row = hb + ci * 1024;
      const float* wr = w + (c * 8 + ci) * 5;
#pragma unroll
      for (int t = 0; t < 5; ++t) {
        int j = base + t;
        float hv = (j >= 0 && j < 1024) ? row[j] : 0.0f;
        s = fmaf(hv, wr[t], s);
      }
    }
    m = fmaxf(m, leaky_f(s));
  }
  h2[idx] = m;
}

// ---------------------------------------------------------------------------
// Kernel 3 (fused, one block per batch b, 256 threads = 8 wave32):
//   a[n,h]  = sum_c h2[b,c,n] * g_w[h,c]        (WMMA f32 16x16x4, K=8)
//   b2[n,h] = sum_c h2[b,c,n] * g_w[h,c+8]
//   rel[b,h] = mean_{i,j} leaky(a[i,h] + b2[j,h] + g_b[h])
// ---------------------------------------------------------------------------
__global__ void __launch_bounds__(256) relational_fused(const float* __restrict__ h2,
                                                        const float* __restrict__ g_w,
                                                        const float* __restrict__ g_b,
                                                        float* __restrict__ rel) {
  __shared__ float hs[8 * 128];     // h2[b] as (c, n)
  __shared__ float gw_s[64 * 16];   // g_w  as (h, 2F)
  __shared__ float gb_s[64];
  __shared__ float a_s[128 * 64];   // a  (n, h)
  __shared__ float b_s[128 * 64];   // b2 (n, h)
  __shared__ float part[4 * 64];

  const int b = blockIdx.x;
  const int tid = threadIdx.x;

  for (int i = tid; i < 8 * 128; i += 256) hs[i] = h2[b * 1024 + i];
  for (int i = tid; i < 64 * 16; i += 256) gw_s[i] = g_w[i];
  if (tid < 64) gb_s[tid] = g_b[tid];
  __syncthreads();

  const int wave = tid >> 5;
  const int lane = tid & 31;
  const int row  = lane & 15;            // M (a-frag) / N (b-frag) within tile
  const int kofs = (lane >> 4) << 1;     // K sub-offset per half-wave

  // 64 output tiles total (8 n-tiles x 4 h-tiles x {a,b2}); 8 per wave.
  // Uniform loop bounds per wave -> EXEC stays all-1s around WMMA.
  for (int t = wave * 8; t < wave * 8 + 8; ++t) {
    const int sel    = t & 1;            // 0 -> a, 1 -> b2
    const int h_tile = (t >> 1) & 3;
    const int m_tile = t >> 3;
    const int m_g = m_tile * 16 + row;
    const int h_g = h_tile * 16 + row;
    v8f acc = {};
#pragma unroll
    for (int s = 0; s < 2; ++s) {        // K = 8 in two K=4 steps
      const int k0 = 4 * s + kofs;
      v2f av, bv;
      av.x = hs[(k0 + 0) * 128 + m_g];   // A[m][k] = h2[c=k, n=m]
      av.y = hs[(k0 + 1) * 128 + m_g];
      const int kb = k0 + 8 * sel;
      bv.x = gw_s[h_g * 16 + kb + 0];    // B[k][h] = g_w[h][k (+8)]
      bv.y = gw_s[h_g * 16 + kb + 1];
      acc = __builtin_amdgcn_wmma_f32_16x16x4_f32(false, av, false, bv,
                                                  (short)0, acc, false, false);
    }
    float* dst = sel ? b_s : a_s;
    const int col = h_tile * 16 + (lane & 15);
    const int rowbase = m_tile * 16 + ((lane >> 4) << 3);
#pragma unroll
    for (int v = 0; v < 8; ++v)
      dst[(rowbase + v) * 64 + col] = acc[v];
  }
  __syncthreads();

  // N^2 leaky reduction: thread -> (h = tid&63, i-quarter = tid>>6).
  // b_s[j*64+h]: fixed h per lane, distinct h across lanes -> distinct banks.
  const int h = tid & 63;
  const int q = tid >> 6;
  const float bb = gb_s[h];
  float acc = 0.0f;
  for (int i = q * 32; i < q * 32 + 32; ++i) {
    const float ai = a_s[i * 64 + h] + bb;
#pragma unroll 4
    for (int j = 0; j < 128; ++j) {
      const float s = ai + b_s[j * 64 + h];
      acc += fmaxf(s, LEAKY * s);
    }
  }
  part[q * 64 + h] = acc;
  __syncthreads();
  if (tid < 64) {
    const float s = part[tid] + part[64 + tid] + part[128 + tid] + part[192 + tid];
    rel[b * 64 + tid] = s * (1.0f / (128.0f * 128.0f));
  }
}

// ---------------------------------------------------------------------------
// Kernel 4: out = leaky(rel @ f_w^T + f_b)   (M=128, N=64, K=64)
// WMMA f32 16x16x4, 16 K-steps. One wave per 16x16 output tile.
// ---------------------------------------------------------------------------
__global__ void __launch_bounds__(128) final_gemm(const float* __restrict__ rel,
                                                  const float* __restrict__ f_w,
                                                  const float* __restrict__ f_b,
                                                  float* __restrict__ out) {
  const int mtile = blockIdx.x;          // 0..7
  const int ntile = threadIdx.x >> 5;    // 0..3
  const int lane  = threadIdx.x & 31;
  const int row   = lane & 15;
  const int kofs  = (lane >> 4) << 1;
  const int m_g = mtile * 16 + row;
  const int n_g = ntile * 16 + row;
  v8f acc = {};
#pragma unroll
  for (int k0 = 0; k0 < 64; k0 += 4) {
    v2f av, bv;
    av.x = rel[m_g * 64 + k0 + kofs + 0];  // A[m][k] = rel[m][k]
    av.y = rel[m_g * 64 + k0 + kofs + 1];
    bv.x = f_w[n_g * 64 + k0 + kofs + 0];  // B[k][n] = f_w[n][k]
    bv.y = f_w[n_g * 64 + k0 + kofs + 1];
    acc = __builtin_amdgcn_wmma_f32_16x16x4_f32(false, av, false, bv,
                                                (short)0, acc, false, false);
  }
  const int col = ntile * 16 + (lane & 15);
  const float bb = f_b[col];
  const int rowbase = mtile * 16 + ((lane >> 4) << 3);
#pragma unroll
  for (int v = 0; v < 8; ++v) {
    const float s = acc[v] + bb;
    out[(rowbase + v) * 64 + col] = fmaxf(s, LEAKY * s);
  }
}

// ---------------------------------------------------------------------------
extern "C" void kernel_launch(void* const* d_in, const int* in_sizes, int n_in,
                              void* d_out, int out_size, void* d_ws, size_t ws_size,
                              hipStream_t stream) {
  const float* x   = (const float*)d_in[0];
  const float* c1w = (const float*)d_in[1];
  const float* c1b = (const float*)d_in[2];
  const float* c2w = (const float*)d_in[3];
  const float* c2b = (const float*)d_in[4];
  const float* gw  = (const float*)d_in[5];
  const float* gb  = (const float*)d_in[6];
  const float* fw  = (const float*)d_in[7];
  const float* fb  = (const float*)d_in[8];
  float* out = (float*)d_out;

  float* ws  = (float*)d_ws;
  float* h1  = ws;                        // 128*8*1024 floats (4 MB)
  float* h2  = h1 + 128 * 8 * 1024;       // 128*8*128 floats (512 KB)
  float* rel = h2 + 128 * 8 * 128;        // 128*64 floats (32 KB)

  conv1_pool4<<<(128 * 8 * 1024) / 256, 256, 0, stream>>>(x, c1w, c1b, h1);
  conv2_pool8<<<(128 * 8 * 128) / 256, 256, 0, stream>>>(h1, c2w, c2b, h2);
  relational_fused<<<128, 256, 0, stream>>>(h2, gw, gb, rel);
  final_gemm<<<8, 128, 0, stream>>>(rel, fw, fb, out);
}